// Net_25477746000039
// MI455X (gfx1250) — compile-verified
//
#include <hip/hip_runtime.h>
#include <hip/hip_bf16.h>

typedef float v2f __attribute__((ext_vector_type(2)));
typedef float v8f __attribute__((ext_vector_type(8)));

#if __has_builtin(__builtin_amdgcn_tanhf)
#define TANH_F32(v) __builtin_amdgcn_tanhf(v)
#else
#define TANH_F32(v) tanhf(v)
#endif

// D = A(16x4 f32) * B(4x16 f32) + C(16x16 f32), full fp32 WMMA.
__device__ __forceinline__ v8f wmma4(v2f a, v2f b, v8f c) {
    return __builtin_amdgcn_wmma_f32_16x16x4_f32(
        false, a, false, b, (short)0, c, false, false);
}

// One wave32 owns a 16-row batch tile. State = H^T held in WMMA D layout:
//   d[r] @ lane L = h[m][n],  m = r + 8*(L>=16),  n = L&15.
// K-slot permutation trick: slice j of the K=16 reduction covers hidden rows
// {2j, 2j+1, 2j+8, 2j+9}. Under the (assumed, self-consistent) interleaved
// slot layout (V0 = slot0/slot2 per lane-half, V1 = slot1/slot3), the B
// operand of slice j is exactly (d[2j], d[2j+1]) — the previous step's D
// output IS the next step's B operand with zero cross-lane movement.
__global__ __launch_bounds__(128) void rnn_wmma_f32_kernel(
    const float* __restrict__ x,        // [B, T]
    const float* __restrict__ h0,       // [B, 16]
    const float* __restrict__ W_ih,     // [16]
    const float* __restrict__ W_hh,     // [16,16] row-major W_hh[m][k]
    const float* __restrict__ b_ih,     // [16]
    const float* __restrict__ b_hh,     // [16]
    const float* __restrict__ W_lin,    // [16]
    const float* __restrict__ b_lin,    // [1]
    float* __restrict__ out,            // [B*T] then [B,16] (hT) appended
    int B, int T)
{
    const int lane = threadIdx.x & 31;
    const int wave = threadIdx.x >> 5;
    const int tile = blockIdx.x * 4 + wave;
    const int b0   = tile * 16;
    if (b0 >= B) return;                 // wave-uniform, EXEC stays full

    const int  n      = lane & 15;
    const bool lolane = lane < 16;
    const int  moff   = lolane ? 0 : 8;

    // ---- constant A operand: W_hh with the permuted K-slot assignment ----
    // slice j, lane L (M = L&15):
    //   a.x = W_hh[m][ lo ? 2j   : 2j+8 ]   (slot0 / slot2)
    //   a.y = W_hh[m][ lo ? 2j+1 : 2j+9 ]   (slot1 / slot3)
    v2f A[4];
#pragma unroll
    for (int j = 0; j < 4; ++j) {
        const int kx = 2 * j + (lolane ? 0 : 8);
        A[j].x = W_hh[n * 16 + kx];
        A[j].y = W_hh[n * 16 + kx + 1];
    }

    // per-hidden-row constants replicated into the D-layout lane pattern
    float wih[8], bias[8], wlin[8];
#pragma unroll
    for (int r = 0; r < 8; ++r) {
        const int m = r + moff;
        wih[r]  = W_ih[m];
        bias[r] = b_ih[m] + b_hh[m];
        wlin[r] = W_lin[m];
    }
    const float blin = b_lin[0];

    // ---- initial hidden state into D layout ----
    float d[8];
    {
        const float* hp = h0 + (size_t)(b0 + n) * 16 + moff;
#pragma unroll
        for (int r = 0; r < 8; ++r) d[r] = hp[r];
    }

    const v8f zeroc = {};

    const size_t row  = (size_t)(b0 + n) * (size_t)T;
    const float* xrow = x + row;
    float*       orow = out + row;

    for (int t0 = 0; t0 < T; t0 += 4) {
        if (t0 + 64 < T) __builtin_prefetch(xrow + t0 + 64, 0, 0);
        const float4 xq = *(const float4*)(xrow + t0);
        const float xv[4] = {xq.x, xq.y, xq.z, xq.w};
        float ov[4];
#pragma unroll
        for (int u = 0; u < 4; ++u) {
            // B operands are just pairs of the current state registers
            v2f B0 = {d[0], d[1]};
            v2f B1 = {d[2], d[3]};
            v2f B2 = {d[4], d[5]};
            v2f B3 = {d[6], d[7]};

            // C seeded with input projection + both biases (xp^T layout == D)
            v8f ca;
#pragma unroll
            for (int r = 0; r < 8; ++r) ca[r] = fmaf(xv[u], wih[r], bias[r]);

            // two independent 2-deep WMMA chains (halved critical latency)
            ca = wmma4(A[0], B0, ca);
            ca = wmma4(A[1], B1, ca);
            v8f cb = wmma4(A[2], B2, zeroc);
            cb = wmma4(A[3], B3, cb);

            // tanh (hardware v_tanh_f32) + linear head partial
            float osum = 0.f;
#pragma unroll
            for (int r = 0; r < 8; ++r) {
                d[r] = TANH_F32(ca[r] + cb[r]);
                osum = fmaf(wlin[r], d[r], osum);
            }
            osum += __shfl_xor(osum, 16, 32);   // combine the two lane-halves
            ov[u] = osum + blin;
        }
        if (lolane)
            *(float4*)(orow + t0) = make_float4(ov[0], ov[1], ov[2], ov[3]);
    }

    // ---- final hidden state hT appended after out[B*T] ----
    float* hT = out + (size_t)B * (size_t)T + (size_t)(b0 + n) * 16 + moff;
#pragma unroll
    for (int r = 0; r < 8; ++r) hT[r] = d[r];
}

extern "C" void kernel_launch(void* const* d_in, const int* in_sizes, int n_in,
                              void* d_out, int out_size, void* d_ws, size_t ws_size,
                              hipStream_t stream) {
    const float* x     = (const float*)d_in[0];   // [B,T,1]
    const float* hprev = (const float*)d_in[1];   // [1,B,16]
    const float* W_ih  = (const float*)d_in[2];   // [16,1]
    const float* W_hh  = (const float*)d_in[3];   // [16,16]
    const float* b_ih  = (const float*)d_in[4];   // [16]
    const float* b_hh  = (const float*)d_in[5];   // [16]
    const float* W_lin = (const float*)d_in[6];   // [1,16]
    const float* b_lin = (const float*)d_in[7];   // [1]
    float* out = (float*)d_out;

    const int B = in_sizes[1] / 16;               // hidden_prev = B*16
    const int T = in_sizes[0] / B;                // x = B*T

    const int tiles  = B / 16;                    // one wave32 per tile
    const int blocks = (tiles + 3) / 4;           // 4 waves (128 thr) / block
    rnn_wmma_f32_kernel<<<blocks, 128, 0, stream>>>(
        x, hprev, W_ih, W_hh, b_ih, b_hh, W_lin, b_lin, out, B, T);
}